// VectorQuantizer_30081950941234
// MI455X (gfx1250) — compile-verified
//
#include <hip/hip_runtime.h>
#include <math.h>

typedef float v2f __attribute__((ext_vector_type(2)));
typedef float v8f __attribute__((ext_vector_type(8)));

#define KCODES 1024
#define DIM    64
#define HW     1024          // H*W
#define NTOT   65536         // B*H*W
#define ZELEMS 4194304       // B*D*H*W
#define BETA   0.25f

// d_out float layout: [0..4194303] z_q_st, [4194304] vq_loss, [4194305] commit,
// [4194306..4259841] indices, [4259842] usage, [4259843] perplexity
#define OUT_VQ    4194304
#define OUT_CM    4194305
#define OUT_IDX   4194306
#define OUT_USE   4259842
#define OUT_PPL   4259843

// d_ws float layout: counts[1024] @0, cnorm[1024] @1024, bsum[1024] @2048,
// ctP[65536] @4096  (k-pair-interleaved codebook transpose:
//                    ctP[p*2048 + col*2 + e] = cb[col][2p + e], p = 0..31)

__global__ void vq_prep(const float* __restrict__ cb, float* __restrict__ ws) {
    int tid = blockIdx.x * blockDim.x + threadIdx.x;   // 0..65535
    int e = tid & 1;
    int c = (tid >> 1) & (KCODES - 1);
    int p = tid >> 11;
    ws[4096 + tid] = cb[c * DIM + 2 * p + e];          // packed B tile layout
    if (tid < KCODES) {
        const float* row = cb + tid * DIM;
        float s = 0.f;
        #pragma unroll
        for (int i = 0; i < DIM; i++) s += row[i] * row[i];
        ws[1024 + tid] = s;    // cnorm
        ws[tid] = 0.f;         // counts (re-init every launch)
    }
}

__global__ __launch_bounds__(128)
void vq_main(const float* __restrict__ z, const float* __restrict__ cb,
             const float* __restrict__ ctP, const float* __restrict__ cnorm,
             float* __restrict__ counts, float* __restrict__ bsum,
             float* __restrict__ out) {
    __shared__ float lds[4][16 * 65];
    __shared__ float wsum[4];

    const int wave = threadIdx.x >> 5;
    const int lane = threadIdx.x & 31;
    const int tile = blockIdx.x * 4 + wave;            // 0..4095
    const int n0   = tile * 16;
    const int b    = n0 >> 10;
    const int hw0  = n0 & (HW - 1);
    const float* zb = z   + (size_t)b * DIM * HW + hw0; // (m,d) at zb[d*HW + m]
    float*       ob = out + (size_t)b * DIM * HW + hw0;

    const int m    = lane & 15;            // row for A / epilogue
    const int koff = (lane >> 4) << 1;     // k-pair select per ISA A-layout
    const int phalf = lane >> 4;           // pair offset: p = 2j + phalf

    // A = -2 * z_tile (16x64), striped per V_WMMA_F32_16X16X4_F32 A layout
    v2f a[16];
    #pragma unroll
    for (int j = 0; j < 16; j++) {
        int k0 = 4 * j + koff;
        a[j].x = -2.0f * zb[k0 * HW + m];
        a[j].y = -2.0f * zb[(k0 + 1) * HW + m];
    }

    float minv[8];
    int   mini[8];
    #pragma unroll
    for (int i = 0; i < 8; i++) { minv[i] = 3.4e38f; mini[i] = 0; }

    // score(m, c) = ||c||^2 - 2 z_m . c  ; sweep 64 column tiles of 16 codes
    for (int c0 = 0; c0 < KCODES; c0 += 16) {
        const int col = c0 + (lane & 15);
        const float cn = cnorm[col];
        v8f acc = {0.f, 0.f, 0.f, 0.f, 0.f, 0.f, 0.f, 0.f};  // inline-0 SRC2
        const float* bp = ctP + (size_t)col * 2 + phalf * 2048;
        #pragma unroll
        for (int j = 0; j < 16; j++) {
            // pair p = 2j + phalf -> one aligned 8B load per lane (global_load_b64)
            v2f bv = *(const v2f*)(bp + (size_t)j * 4096);
            acc = __builtin_amdgcn_wmma_f32_16x16x4_f32(
                false, a[j], false, bv, (short)0, acc, false, false);
        }
        #pragma unroll
        for (int i = 0; i < 8; i++) {
            float sc = acc[i] + cn;
            if (sc < minv[i]) { minv[i] = sc; mini[i] = col; }
        }
    }

    // argmin across the 16 lanes holding each row (C layout: slot i = row i (+8 hi half))
    #pragma unroll
    for (int off = 1; off < 16; off <<= 1) {
        #pragma unroll
        for (int i = 0; i < 8; i++) {
            float ov = __shfl_xor(minv[i], off, 32);
            int   oi = __shfl_xor(mini[i], off, 32);
            if (ov < minv[i] || (ov == minv[i] && oi < mini[i])) {
                minv[i] = ov; mini[i] = oi;
            }
        }
    }

    // broadcast winners to all lanes: rows 0..7 from lane 0, rows 8..15 from lane 16
    int idxr[16];
    #pragma unroll
    for (int i = 0; i < 8; i++) {
        idxr[i]     = __shfl(mini[i], 0, 32);
        idxr[i + 8] = __shfl(mini[i], 16, 32);
    }

    // indices (as float) + histogram (float adds of 1.0 are exact & order-independent)
    if (lane == 0 || lane == 16) {
        const int rbase = (lane >> 4) * 8;
        #pragma unroll
        for (int i = 0; i < 8; i++) {
            int idx = mini[i];
            out[(size_t)OUT_IDX + n0 + rbase + i] = (float)idx;
            atomicAdd(&counts[idx], 1.0f);
        }
    }

    // stage the 16 winning codebook rows to LDS (stride 65 -> conflict-free gather)
    #pragma unroll
    for (int mm = 0; mm < 16; mm++) {
        const float* crow = cb + (size_t)idxr[mm] * DIM + 2 * lane;
        lds[wave][mm * 65 + 2 * lane]     = crow[0];
        lds[wave][mm * 65 + 2 * lane + 1] = crow[1];
    }

    // epilogue: z_q_st (== z_q in value) + squared-diff sum; d-major => coalesced
    float lsum = 0.f;
    const int dbase = (lane >> 4) * 32;
    #pragma unroll
    for (int j = 0; j < 32; j++) {
        const int d = dbase + j;
        float zq = lds[wave][m * 65 + d];
        float ze = zb[d * HW + m];
        float df = zq - ze;
        lsum += df * df;
        ob[d * HW + m] = zq;
    }

    #pragma unroll
    for (int off = 16; off > 0; off >>= 1) lsum += __shfl_xor(lsum, off, 32);
    if (lane == 0) wsum[wave] = lsum;
    __syncthreads();
    if (threadIdx.x == 0)
        bsum[blockIdx.x] = (wsum[0] + wsum[1]) + (wsum[2] + wsum[3]);
}

__global__ __launch_bounds__(1024)
void vq_final(const float* __restrict__ counts, const float* __restrict__ bsum,
              float* __restrict__ out) {
    __shared__ float s_ent[1024], s_use[1024], s_loss[1024];
    const int t = threadIdx.x;
    float c = counts[t];
    float p = c * (1.0f / (float)NTOT);
    s_ent[t]  = p * logf(p + 1e-10f);
    s_use[t]  = (c > 0.f) ? 1.f : 0.f;
    s_loss[t] = bsum[t];
    __syncthreads();
    for (int off = 512; off > 0; off >>= 1) {
        if (t < off) {
            s_ent[t]  += s_ent[t + off];
            s_use[t]  += s_use[t + off];
            s_loss[t] += s_loss[t + off];
        }
        __syncthreads();
    }
    if (t == 0) {
        float vq = s_loss[0] * (1.0f / (float)ZELEMS);
        out[OUT_VQ]  = vq;
        out[OUT_CM]  = BETA * vq;
        out[OUT_USE] = s_use[0] * (1.0f / (float)KCODES);
        out[OUT_PPL] = expf(-s_ent[0]);
    }
}

extern "C" void kernel_launch(void* const* d_in, const int* in_sizes, int n_in,
                              void* d_out, int out_size, void* d_ws, size_t ws_size,
                              hipStream_t stream) {
    (void)in_sizes; (void)n_in; (void)out_size; (void)ws_size;
    const float* z  = (const float*)d_in[0];   // (64,64,32,32) f32
    const float* cb = (const float*)d_in[1];   // (1024,64) f32
    float* ws     = (float*)d_ws;
    float* counts = ws;            // 1024
    float* cnorm  = ws + 1024;     // 1024
    float* bsum   = ws + 2048;     // 1024
    float* ctP    = ws + 4096;     // 65536 (pair-interleaved codebook transpose)
    float* out    = (float*)d_out;

    vq_prep <<<256, 256, 0, stream>>>(cb, ws);
    vq_main <<<1024, 128, 0, stream>>>(z, cb, ctP, cnorm, counts, bsum, out);
    vq_final<<<1, 1024, 0, stream>>>(counts, bsum, out);
}